// ModelNew_32753420600123
// MI455X (gfx1250) — compile-verified
//
#include <hip/hip_runtime.h>

typedef __attribute__((ext_vector_type(2))) float v2f;
typedef __attribute__((ext_vector_type(8))) float v8f;

#define D_MODEL 1024
#define D_MEM 256
#define N_HEADS 16
#define D_KEY_HEAD 96
#define D_MODEL_HEAD 64
#define BATCH 4
#define SEQ 8192

#define WAVES 4
#define ROWS_PER_BLOCK (WAVES * 16)   // 64
#define KCHUNK 32
#define WMQ_PSTR 288                  // pair rows: 128 cols*2 + 32 pad (dwords)
#define QTA_STRIDE 132                // 128 + 4 pad
#define F_STRIDE 98                   // 96 + 2 pad (even for b64 A-frags)

#define QTA_FLOATS (16 * QTA_STRIDE)             // 2112
#define FTS_FLOATS (16 * F_STRIDE)               // 1568
#define WAVE_FLOATS (QTA_FLOATS + FTS_FLOATS + 16)  // 3696
#define SMEM_FLOATS (WAVES * WAVE_FLOATS)        // 14784 floats = 59136 B
// phase-1 Wmq staging region: 16*WMQ_PSTR = 4608 floats at smem[0],
// time-overlaps wave 0/1 phase-2 regions (guarded by barriers)

__global__ __launch_bounds__(WAVES * 32, 1)
void fastweight_dpfp_kernel(const float* __restrict__ hidden,
                            const float* __restrict__ Wmq,
                            const float* __restrict__ Wmem,
                            const float* __restrict__ z,
                            float* __restrict__ out) {
  __shared__ __align__(16) float smem[SMEM_FLOATS];

  const int tid  = threadIdx.x;
  const int wave = tid >> 5;
  const int lane = tid & 31;
  const int half = lane >> 4;
  const int l16  = lane & 15;

  const int g0   = blockIdx.x * ROWS_PER_BLOCK;
  const int b    = g0 / SEQ;
  const int wrow = g0 + wave * 16;

  const float* __restrict__ hidRow = hidden + (size_t)wrow * D_MODEL;
  const float* __restrict__ zb     = z    + (size_t)b * N_HEADS * D_KEY_HEAD;
  const float* __restrict__ wmemB  = Wmem + (size_t)b * N_HEADS * D_KEY_HEAD * D_MODEL_HEAD;
  float* __restrict__ outRow = out + (size_t)wrow * D_MODEL;

  float* qtw = &smem[wave * WAVE_FLOATS];   // [16 rows][QTA_STRIDE]
  float* fts = qtw + QTA_FLOATS;            // [16 rows][F_STRIDE]
  float* dnm = fts + FTS_FLOATS;            // [16]

  for (int pass = 0; pass < 2; ++pass) {    // heads 0-7, then 8-15
    v8f acc[8] = {};

    // -------- Phase 1: q[:, pass*128 .. +128) = hidden @ Wmq --------------
    for (int kc = 0; kc < D_MODEL; kc += KCHUNK) {
      const float* srcW = Wmq + (size_t)kc * D_MEM + pass * 128;
      for (int i = tid; i < 16 * 32; i += WAVES * 32) {
        int p  = i >> 5;
        int c4 = (i & 31) * 4;
        float4 a4 = *(const float4*)(srcW + (size_t)(2 * p)     * D_MEM + c4);
        float4 b4 = *(const float4*)(srcW + (size_t)(2 * p + 1) * D_MEM + c4);
        float2* dst = (float2*)&smem[p * WMQ_PSTR + c4 * 2];
        dst[0] = make_float2(a4.x, b4.x);
        dst[1] = make_float2(a4.y, b4.y);
        dst[2] = make_float2(a4.z, b4.z);
        dst[3] = make_float2(a4.w, b4.w);
      }
      __syncthreads();
#pragma unroll
      for (int ks = 0; ks < KCHUNK; ks += 4) {
        const float* ap = hidRow + (size_t)l16 * D_MODEL + (kc + ks + half * 2);
        float2 av = *(const float2*)ap;
        v2f a; a.x = av.x; a.y = av.y;
        const int p0 = (ks >> 1) + half;
#pragma unroll
        for (int t = 0; t < 8; ++t) {
          float2 bv = *(const float2*)&smem[p0 * WMQ_PSTR + (t * 16 + l16) * 2];
          v2f bf; bf.x = bv.x; bf.y = bv.y;
          acc[t] = __builtin_amdgcn_wmma_f32_16x16x4_f32(
              false, a, false, bf, (short)0, acc[t], false, false);
        }
      }
      __syncthreads();
    }

    // dump all 8 q tiles (WMMA C layout -> row-major) into per-wave LDS
#pragma unroll
    for (int t = 0; t < 8; ++t)
#pragma unroll
      for (int v = 0; v < 8; ++v)
        qtw[(v + 8 * half) * QTA_STRIDE + t * 16 + l16] = acc[t][v];
    // acc dead from here; per-wave LDS, in-wave DS ordering suffices

    // -------- Phase 2: DPFP + out = feats @ Wmem / denom ------------------
    for (int hh = 0; hh < 8; ++hh) {        // dynamic loop (small code)
      const int h = pass * 8 + hh;

      float x[32];
#pragma unroll
      for (int i = 0; i < 16; ++i) {
        float qv = qtw[l16 * QTA_STRIDE + hh * 16 + i];
        x[i]      = qv > 0.f ? qv : 0.f;    // relu(q)
        x[16 + i] = qv < 0.f ? -qv : 0.f;   // relu(-q)
      }
      float f[48];
      float ss = 0.f;
      if (half == 0) {                      // K = 0..47 (all indices static)
#pragma unroll
        for (int kk = 0; kk < 48; ++kk) {
          const int j = kk >> 5, i = kk & 31;
          float v = x[i] * x[(i - (j + 1)) & 31];
          f[kk] = v; ss += v * v;
        }
      } else {                              // K = 48..95
#pragma unroll
        for (int kk = 0; kk < 48; ++kk) {
          const int K = 48 + kk, j = K >> 5, i = K & 31;
          float v = x[i] * x[(i - (j + 1)) & 31];
          f[kk] = v; ss += v * v;
        }
      }
      ss += __shfl_xor(ss, 16, 32);
      const float inv = rsqrtf(ss + 1e-12f);
      float dd = 0.f;
      if (half == 0) {
#pragma unroll
        for (int kk = 0; kk < 48; ++kk) {
          float v = f[kk] * inv;
          fts[l16 * F_STRIDE + kk] = v;
          dd += v * zb[h * D_KEY_HEAD + kk];
        }
      } else {
#pragma unroll
        for (int kk = 0; kk < 48; ++kk) {
          float v = f[kk] * inv;
          fts[l16 * F_STRIDE + 48 + kk] = v;
          dd += v * zb[h * D_KEY_HEAD + 48 + kk];
        }
      }
      dd += __shfl_xor(dd, 16, 32);
      if (half == 0) dnm[l16] = 1e-5f + dd;

      // GEMM2: out(16x64) = feats(16x96) @ Wmem[b][h](96x64); B from global
      const float* __restrict__ wmemH =
          wmemB + (size_t)h * (D_KEY_HEAD * D_MODEL_HEAD);
      v8f oc[4] = {};
#pragma unroll
      for (int kk = 0; kk < D_KEY_HEAD; kk += 4) {
        float2 av = *(const float2*)&fts[l16 * F_STRIDE + kk + half * 2];
        v2f a; a.x = av.x; a.y = av.y;
        const float* bp = wmemH + (kk + half * 2) * D_MODEL_HEAD + l16;
#pragma unroll
        for (int t = 0; t < 4; ++t) {
          v2f bf; bf.x = bp[t * 16]; bf.y = bp[t * 16 + D_MODEL_HEAD];
          oc[t] = __builtin_amdgcn_wmma_f32_16x16x4_f32(
              false, a, false, bf, (short)0, oc[t], false, false);
        }
      }

      float invd[8];
#pragma unroll
      for (int v = 0; v < 8; ++v) invd[v] = 1.0f / dnm[v + 8 * half];
#pragma unroll
      for (int t = 0; t < 4; ++t)
#pragma unroll
        for (int v = 0; v < 8; ++v) {
          int M = v + 8 * half;
          outRow[(size_t)M * D_MODEL + h * D_MODEL_HEAD + t * 16 + l16] =
              oc[t][v] * invd[v];
        }
    }
    __syncthreads();  // pass boundary: Wmq staging overlaps wave LDS regions
  }
}

extern "C" void kernel_launch(void* const* d_in, const int* in_sizes, int n_in,
                              void* d_out, int out_size, void* d_ws, size_t ws_size,
                              hipStream_t stream) {
  (void)in_sizes; (void)n_in; (void)out_size; (void)d_ws; (void)ws_size;
  const float* hidden = (const float*)d_in[0];
  const float* Wmq    = (const float*)d_in[1];
  const float* Wmem   = (const float*)d_in[2];
  const float* z      = (const float*)d_in[3];
  float* out          = (float*)d_out;

  dim3 grid((BATCH * SEQ) / ROWS_PER_BLOCK);  // 512 blocks
  dim3 block(WAVES * 32);                     // 128 threads = 4 waves
  hipLaunchKernelGGL(fastweight_dpfp_kernel, grid, block, 0, stream,
                     hidden, Wmq, Wmem, z, out);
}